// LSTMModel_25598005084838
// MI455X (gfx1250) — compile-verified
//
#include <hip/hip_runtime.h>

typedef float v2f __attribute__((ext_vector_type(2)));
typedef float v8f __attribute__((ext_vector_type(8)));

#define T_STEPS 4096
#define BATCH   4096
#define IN_DIM  5
#define HID     10

// wave32 half-swap: exchange lane n <-> lane n+16 (ds_swizzle group-of-32, xor=0x10, and=0x1f)
__device__ __forceinline__ float swap16(float x) {
  int i = __builtin_bit_cast(int, x);
  i = __builtin_amdgcn_ds_swizzle(i, 0x401F);
  return __builtin_bit_cast(float, i);
}

__device__ __forceinline__ float raw_exp2(float x) {
#if __has_builtin(__builtin_amdgcn_exp2f)
  return __builtin_amdgcn_exp2f(x);   // raw v_exp_f32, no range fixup
#else
  return __builtin_exp2f(x);
#endif
}

#if __has_builtin(__builtin_amdgcn_tanhf)
// native v_tanh_f32 (CDNA5 transcendental)
__device__ __forceinline__ float ftanh(float x) { return __builtin_amdgcn_tanhf(x); }
__device__ __forceinline__ float fsig(float x) {
  return __builtin_fmaf(0.5f, __builtin_amdgcn_tanhf(0.5f * x), 0.5f);
}
#else
__device__ __forceinline__ float ftanh(float x) {
  float e = raw_exp2(x * -2.8853900817779268f);     // exp(-2x)
  return (1.0f - e) * __builtin_amdgcn_rcpf(1.0f + e);
}
__device__ __forceinline__ float fsig(float x) {
  return __builtin_amdgcn_rcpf(1.0f + raw_exp2(x * -1.4426950408889634f));
}
#endif

// One wave handles 16 batch columns for all 4096 timesteps.
// Gates G(48x16) = Wfull(48x16) @ Z(16x16), Z = [x_t(5) ; h(10) ; 0],
// computed as 3 M-tiles x 4 K-chunks of V_WMMA_F32_16X16X4_F32 with the
// weight matrix resident in the A-operand VGPRs for the whole kernel.
__global__ __launch_bounds__(32) void lstm_wmma_kernel(
    const float* __restrict__ x,     // (T,B,5)
    const float* __restrict__ W_ih,  // (40,5)
    const float* __restrict__ W_hh,  // (40,10)
    const float* __restrict__ b_ih,  // (40)
    const float* __restrict__ b_hh,  // (40)
    const float* __restrict__ W_lab, // (5,10)
    const float* __restrict__ b_lab, // (5)
    float* __restrict__ out)         // (B,5)
{
  const int  lane = threadIdx.x;   // 0..31
  const int  n    = lane & 15;     // batch column within tile
  const bool hi   = lane >= 16;    // lane half
  const int  base = blockIdx.x * 16;

  // ---- A operand (weights), WMMA A-layout, stationary for entire kernel ----
  // A[m][q] lane L: row = 16m + (L&15), k = 4q + 2*hi + v ; value Wfull(row,k)
  v2f A[3][4];
  #pragma unroll
  for (int m = 0; m < 3; ++m)
    #pragma unroll
    for (int q = 0; q < 4; ++q)
      #pragma unroll
      for (int v = 0; v < 2; ++v) {
        int row = 16 * m + n;
        int k   = 4 * q + (hi ? 2 : 0) + v;
        float val = 0.0f;
        if (row < 40) {
          if (k < IN_DIM)            val = W_ih[row * IN_DIM + k];
          else if (k < IN_DIM + HID) val = W_hh[row * HID + (k - IN_DIM)];
        }
        A[m][q][v] = val;
      }

  // ---- combined bias in C/D layout: VGPR r holds row 16m + r + 8*hi ----
  float bc[3][8];
  #pragma unroll
  for (int m = 0; m < 3; ++m)
    #pragma unroll
    for (int r = 0; r < 8; ++r) {
      int row = 16 * m + r + (hi ? 8 : 0);
      bc[m][r] = (row < 40) ? (b_ih[row] + b_hh[row]) : 0.0f;
    }

  // lane-private state: lo lanes hold h/c[0..4], hi lanes hold h/c[5..9] of batch n
  float hv[5] = {0.f, 0.f, 0.f, 0.f, 0.f};
  float cv[5] = {0.f, 0.f, 0.f, 0.f, 0.f};

  const int i0 = hi ? 2 : 0;
  const int i1 = hi ? 3 : 1;
  const size_t xstep = (size_t)BATCH * IN_DIM;
  const float* xp = x + (size_t)(base + n) * IN_DIM;  // row t=0
  float xa = xp[i0];
  float xb = xp[i1];
  float xc = xp[4];          // K row 4 (used by lo half only)
  xp += xstep;               // xp -> row min(t+1, T-1) from here on

  for (int t = 0; t < T_STEPS; ++t) {
    // h redistribution for B operand: lo needs h7,h8 (live on hi), hi needs h1,h2 (live on lo)
    float u  = swap16(hi ? hv[2] : hv[1]); // lo <- h7 , hi <- h1
    float vv = swap16(hi ? hv[3] : hv[2]); // lo <- h8 , hi <- h2

    // ---- B operand (Z = [x;h;0]) in WMMA B-layout: K = 4q + 2*hi + v ----
    v2f Bq[4];
    Bq[0][0] = xa;                    // k=0 / k=2  (x)
    Bq[0][1] = xb;                    // k=1 / k=3  (x)
    Bq[1][0] = hi ? u     : xc;       // k=6->h1 / k=4->x4
    Bq[1][1] = hi ? vv    : hv[0];    // k=7->h2 / k=5->h0
    Bq[2][0] = hi ? hv[0] : hv[3];    // k=10->h5 / k=8->h3
    Bq[2][1] = hi ? hv[1] : hv[4];    // k=11->h6 / k=9->h4
    Bq[3][0] = hi ? hv[4] : u;        // k=14->h9 / k=12->h7
    Bq[3][1] = hi ? 0.0f  : vv;       // k=15->0  / k=13->h8

    // prefetch next timestep's x while WMMAs run (clamped: re-reads row T-1 on last step)
    float nxa = xp[i0];
    float nxb = xp[i1];
    float nxc = xp[4];
    if (t + 2 < T_STEPS) xp += xstep;

    // ---- gates = Wfull @ Z + bias : 12x v_wmma_f32_16x16x4_f32 ----
    v8f acc[3];
    #pragma unroll
    for (int m = 0; m < 3; ++m) {
      v8f c;
      #pragma unroll
      for (int r = 0; r < 8; ++r) c[r] = bc[m][r];
      #pragma unroll
      for (int q = 0; q < 4; ++q)
        c = __builtin_amdgcn_wmma_f32_16x16x4_f32(false, A[m][q], false, Bq[q],
                                                  (short)0, c, false, false);
      acc[m] = c;
    }

    // cross-half copies so each lane sees all 48 gate rows of its batch column
    // (compiler dead-codes the ones never read: 17 remain live)
    float swp[3][8];
    #pragma unroll
    for (int m = 0; m < 3; ++m)
      #pragma unroll
      for (int r = 0; r < 8; ++r) swp[m][r] = swap16(acc[m][r]);

    // ---- elementwise LSTM update (torch gate order i,f,g,o) ----
    if (!hi) {
      // lanes 0..15: hidden units j = 0..4
      #pragma unroll
      for (int j = 0; j < 5; ++j) {
        float ig = acc[0][j];                            // rows 0..4   (local)
        float fg = swp[0][j + 2];                        // rows 10..14 (remote)
        float gg = (j < 4) ? acc[1][4 + j] : swp[1][0];  // rows 20..24
        float og;
        if (j == 0)      og = swp[1][6];                 // row 30
        else if (j == 1) og = swp[1][7];                 // row 31
        else             og = acc[2][j - 2];             // rows 32..34
        float is = fsig(ig), fs = fsig(fg), gt = ftanh(gg), os = fsig(og);
        cv[j] = fs * cv[j] + is * gt;
        hv[j] = os * ftanh(cv[j]);
      }
    } else {
      // lanes 16..31: hidden units j = 5..9 (local slot jj = j-5)
      #pragma unroll
      for (int jj = 0; jj < 5; ++jj) {
        float ig = (jj < 3) ? swp[0][5 + jj] : acc[0][jj - 3];   // rows 5..9
        float fg = (jj == 0) ? acc[0][7] : swp[1][jj - 1];       // rows 15..19
        float gg = acc[1][1 + jj];                               // rows 25..29
        float og = swp[2][3 + jj];                               // rows 35..39
        float is = fsig(ig), fs = fsig(fg), gt = ftanh(gg), os = fsig(og);
        cv[jj] = fs * cv[jj] + is * gt;
        hv[jj] = os * ftanh(cv[jj]);
      }
    }

    xa = nxa; xb = nxb; xc = nxc;
  }

  // ---- head: out[b][o] = sum_j h[b][j] * W_lab[o][j] + b_lab[o] ----
  #pragma unroll
  for (int o = 0; o < 5; ++o) {
    float p = 0.0f;
    #pragma unroll
    for (int jj = 0; jj < 5; ++jj)
      p += hv[jj] * W_lab[o * HID + (hi ? 5 : 0) + jj];
    p += swap16(p);                       // combine the two half-sums per batch column
    if (!hi)
      out[(size_t)(base + n) * IN_DIM + o] = p + b_lab[o];
  }
}

extern "C" void kernel_launch(void* const* d_in, const int* in_sizes, int n_in,
                              void* d_out, int out_size, void* d_ws, size_t ws_size,
                              hipStream_t stream) {
  (void)in_sizes; (void)n_in; (void)d_ws; (void)ws_size; (void)out_size;
  const float* x     = (const float*)d_in[0];
  const float* W_ih  = (const float*)d_in[1];
  const float* W_hh  = (const float*)d_in[2];
  const float* b_ih  = (const float*)d_in[3];
  const float* b_hh  = (const float*)d_in[4];
  const float* W_lab = (const float*)d_in[5];
  const float* b_lab = (const float*)d_in[6];
  float* out = (float*)d_out;

  dim3 grid(BATCH / 16);  // 256 independent waves, one 16-column batch tile each
  dim3 block(32);         // one wave32
  lstm_wmma_kernel<<<grid, block, 0, stream>>>(x, W_ih, W_hh, b_ih, b_hh,
                                               W_lab, b_lab, out);
}